// FastTileCoding_1511828488616
// MI455X (gfx1250) — compile-verified
//
#include <hip/hip_runtime.h>

// FastTileCoding for MI455X (gfx1250).
//
// Reference: per sample (3 floats), for 8 tilings x 3 dims, bucketize into a
// UNIFORM 64-bin grid (bins are linspace -> searchsorted == closed-form
// clamp(floor((s-lo)/step),0,63)), gather 8 weights from an 8MB fp32 table,
// sum -> out[i].
//
// Roofline: 2M samples * (12B in + 4B out) + 8MB table (L2-resident) ~= 40MB
// of HBM traffic -> ~1.7us at 23.3 TB/s. Memory-bound; VALU is negligible.
// No matmul structure -> WMMA does not apply. The applicable CDNA5 path is
// the async global->LDS copy (ASYNCcnt + s_wait_asynccnt) used to stage the
// 24 (lo, 1/step) bin descriptors into LDS.

#define NT 8      // tilings
#define ND 3      // dims
#define NB 64     // bins
#define NPAIR (NT * ND)  // 24 (t,d) pairs

#ifndef __has_builtin
#define __has_builtin(x) 0
#endif

#if __has_builtin(__builtin_amdgcn_global_load_async_to_lds_b32) && \
    __has_builtin(__builtin_amdgcn_s_wait_asynccnt)
#define USE_ASYNC_LDS 1
#else
#define USE_ASYNC_LDS 0
#endif

#if USE_ASYNC_LDS
// Builtin signature (per hipcc diagnostic) takes typed int pointers:
//   arg0: global (addrspace(1)) int*   -- source
//   arg1: LDS    (addrspace(3)) int*   -- destination
typedef __attribute__((address_space(1))) int gas_int;
typedef __attribute__((address_space(3))) int las_int;
#endif

__device__ __forceinline__ int bin_index(float s, float lo, float inv_step) {
  // (int)-truncation + clamp == floor + clamp for all inputs.
  float f = (s - lo) * inv_step;
  int i = (int)f;
  i = (i < 0) ? 0 : i;
  i = (i > NB - 1) ? (NB - 1) : i;
  return i;
}

__global__ void __launch_bounds__(256)
tile_coding_kernel(const float* __restrict__ state,     // [n,3]
                   const float* __restrict__ weights,   // [8,64,64,64] flat
                   const float* __restrict__ bins,      // [8,3,65] flat
                   float* __restrict__ out,             // [n]
                   int n) {
  // Stage bins[k*65+0] and bins[k*65+1] (k = t*3+d) into LDS, then derive
  // lo and 1/step per (t,d). These are wave-uniform broadcast reads later.
  __shared__ float s_pair[NPAIR * 2];
  __shared__ float s_lo[NPAIR];
  __shared__ float s_inv[NPAIR];

  const int lt = (int)threadIdx.x;

#if USE_ASYNC_LDS
  if (lt < NPAIR) {
    const float* gp = bins + lt * (NB + 1);
    // Two b32 async copies (260-byte row stride -> only 4B aligned, so no b64).
    // Cast chain: generic int* first (drops const, retypes), then AS-only cast.
    __builtin_amdgcn_global_load_async_to_lds_b32(
        (gas_int*)(int*)(void*)(gp + 0),
        (las_int*)(int*)(void*)&s_pair[2 * lt + 0], 0, 0);
    __builtin_amdgcn_global_load_async_to_lds_b32(
        (gas_int*)(int*)(void*)(gp + 1),
        (las_int*)(int*)(void*)&s_pair[2 * lt + 1], 0, 0);
  }
  __builtin_amdgcn_s_wait_asynccnt(0);  // drain ASYNCcnt (NOP for other waves)
  __syncthreads();
#else
  if (lt < NPAIR) {
    const float* gp = bins + lt * (NB + 1);
    s_pair[2 * lt + 0] = gp[0];
    s_pair[2 * lt + 1] = gp[1];
  }
  __syncthreads();
#endif

  if (lt < NPAIR) {
    float lo = s_pair[2 * lt + 0];
    float step = s_pair[2 * lt + 1] - lo;
    s_lo[lt]  = lo;
    s_inv[lt] = 1.0f / step;
  }
  __syncthreads();

  const int gid0    = (int)(blockIdx.x * blockDim.x + threadIdx.x);
  const int gstride = (int)(gridDim.x * blockDim.x);

  for (int i = gid0; i < n; i += gstride) {
    // 12B coalesced read (compiler merges into a b96/b64+b32 load).
    const float* sp = state + 3 * (long)i;
    const float x0 = sp[0];
    const float x1 = sp[1];
    const float x2 = sp[2];

    float sum = 0.0f;
#pragma unroll
    for (int t = 0; t < NT; ++t) {
      const int k  = t * ND;
      const int i0 = bin_index(x0, s_lo[k + 0], s_inv[k + 0]);
      const int i1 = bin_index(x1, s_lo[k + 1], s_inv[k + 1]);
      const int i2 = bin_index(x2, s_lo[k + 2], s_inv[k + 2]);
      // strides = [64^2, 64, 1], tiling offset t*64^3 = t<<18.
      const int flat = (t << 18) | (i0 << 12) | (i1 << 6) | i2;
      sum += weights[flat];  // 8 independent gathers -> L2-resident, MLP'd
    }
    out[i] = sum;
  }
}

extern "C" void kernel_launch(void* const* d_in, const int* in_sizes, int n_in,
                              void* d_out, int out_size, void* d_ws, size_t ws_size,
                              hipStream_t stream) {
  const float* state   = (const float*)d_in[0];  // [n,3] fp32
  const float* weights = (const float*)d_in[1];  // [8*64^3] fp32
  const float* bins    = (const float*)d_in[2];  // [8*3*65] fp32
  float* out = (float*)d_out;                    // [n] fp32

  const int n = out_size;  // one output per sample
  const int threads = 256; // 8 wave32s per block
  int blocks = (n + threads - 1) / threads;
  if (blocks > 8192) blocks = 8192;  // grid-stride beyond this
  if (blocks < 1) blocks = 1;

  tile_coding_kernel<<<blocks, threads, 0, stream>>>(state, weights, bins, out, n);
}